// PointNet2SAModuleCUDA_5007931867461
// MI455X (gfx1250) — compile-verified
//
#include <hip/hip_runtime.h>
#include <hip/hip_bf16.h>
#include <math.h>

// ---------------------------------------------------------------------------
// PointNet++ SA module for MI455X (gfx1250, wave32).
// Heavy op: e = nfn @ a  (262144 x 212) @ (212 x 212) = 23.6 GFLOP fp32.
// nfn (would be 222 MB) is never materialized to HBM: each sampled point's
// 32x212 tile is built in LDS and consumed by V_WMMA_F32_16X16X4_F32.
// ---------------------------------------------------------------------------

#define NPTS    4096
#define NFPS    1024
#define BATCH   8
#define DFEAT   64
#define NSAMP   32
#define NC      212      // enhanced feature channels
#define COUT    128
#define NFPITCH 216      // padded row pitch for nfn tile in LDS (even)
#define NTILES  14       // ceil(212/16)
#define KSTEPS  53       // 212 / 4 (K of f32 WMMA is 4)

typedef __attribute__((ext_vector_type(2))) float v2f;
typedef __attribute__((ext_vector_type(8))) float v8f;

// ---------------------------------------------------------------- FPS ------
__global__ __launch_bounds__(1024) void fps_kernel(const float* __restrict__ xyz,
                                                   int* __restrict__ fps_idx) {
  const int b   = blockIdx.x;
  const int tid = threadIdx.x;
  const float* X = xyz + (size_t)b * NPTS * 3;

  float px[4], py[4], pz[4], dmin[4];
  for (int i = 0; i < 4; ++i) {
    int nn = tid + i * 1024;
    px[i] = X[nn * 3 + 0];
    py[i] = X[nn * 3 + 1];
    pz[i] = X[nn * 3 + 2];
    dmin[i] = 1e10f;
  }

  __shared__ float s_val[32];
  __shared__ int   s_idx[32];
  __shared__ int   s_cur;
  if (tid == 0) { fps_idx[b * NFPS] = 0; s_cur = 0; }
  __syncthreads();

  const int lane = tid & 31, wid = tid >> 5;
  for (int it = 1; it < NFPS; ++it) {
    const int cur = s_cur;
    const float lx = X[cur * 3 + 0], ly = X[cur * 3 + 1], lz = X[cur * 3 + 2];
    float bv = -1.0f; int bi = 0;
    for (int i = 0; i < 4; ++i) {
      float dx = px[i] - lx, dy = py[i] - ly, dz = pz[i] - lz;
      float d = dx * dx + dy * dy + dz * dz;
      dmin[i] = fminf(dmin[i], d);
      if (dmin[i] > bv) { bv = dmin[i]; bi = tid + i * 1024; }   // first-max
    }
    // wave32 argmax (ties -> smallest index, matches jnp.argmax)
    for (int o = 16; o > 0; o >>= 1) {
      float ov = __shfl_down(bv, o);
      int   oi = __shfl_down(bi, o);
      if (ov > bv || (ov == bv && oi < bi)) { bv = ov; bi = oi; }
    }
    if (lane == 0) { s_val[wid] = bv; s_idx[wid] = bi; }
    __syncthreads();
    if (wid == 0) {
      float v = s_val[lane]; int ix = s_idx[lane];
      for (int o = 16; o > 0; o >>= 1) {
        float ov = __shfl_down(v, o);
        int   oi = __shfl_down(ix, o);
        if (ov > v || (ov == v && oi < ix)) { v = ov; ix = oi; }
      }
      if (lane == 0) { s_cur = ix; fps_idx[b * NFPS + it] = ix; }
    }
    __syncthreads();
  }
}

// --------------------------------------------------------- gather new_xyz --
__global__ void gather_newxyz_kernel(const float* __restrict__ xyz,
                                     const int* __restrict__ fps_idx,
                                     float* __restrict__ out) {
  int t = blockIdx.x * blockDim.x + threadIdx.x;
  if (t >= BATCH * NFPS * 3) return;
  int c = t % 3, pt = t / 3, n = pt % NFPS, b = pt / NFPS;
  out[t] = xyz[((size_t)b * NPTS + fps_idx[b * NFPS + n]) * 3 + c];
}

// ---------------------------------------------------------- ball query -----
__global__ __launch_bounds__(256) void ballquery_kernel(const float* __restrict__ xyz,
                                                        const float* __restrict__ new_xyz,
                                                        int* __restrict__ nb_idx) {
  __shared__ float sx[NPTS], sy[NPTS], sz[NPTS];
  const int b  = blockIdx.x >> 2;        // 4 blocks per batch
  const int n0 = (blockIdx.x & 3) * 256;
  const float* X = xyz + (size_t)b * NPTS * 3;
  for (int j = threadIdx.x; j < NPTS; j += 256) {
    sx[j] = X[j * 3 + 0]; sy[j] = X[j * 3 + 1]; sz[j] = X[j * 3 + 2];
  }
  __syncthreads();

  const int n = n0 + threadIdx.x;
  const float* q = new_xyz + ((size_t)b * NFPS + n) * 3;
  const float qx = q[0], qy = q[1], qz = q[2];
  const float qq = qx * qx + qy * qy + qz * qz;
  int* row = nb_idx + ((size_t)b * NFPS + n) * NSAMP;
  int cnt = 0;
  const float r2 = 0.2f * 0.2f;
  // same expansion as the reference: |q|^2 + |p|^2 - 2 q.p ; first 32 in order
  for (int j = 0; j < NPTS && cnt < NSAMP; ++j) {
    float pp = sx[j] * sx[j] + sy[j] * sy[j] + sz[j] * sz[j];
    float d2 = qq + pp - 2.0f * (qx * sx[j] + qy * sy[j] + qz * sz[j]);
    if (d2 <= r2) row[cnt++] = j;
  }
  int first = (cnt > 0) ? row[0] : (NPTS - 1);
  for (int k = cnt; k < NSAMP; ++k) row[k] = first;
}

// --------------------------- fused feature_enhance + hetero_attention ------
__global__ __launch_bounds__(128) void attn_kernel(
    const float* __restrict__ xyz, const float* __restrict__ features,
    const float* __restrict__ Amat, const float* __restrict__ W1,
    const float* __restrict__ B1, const int* __restrict__ fps_idx,
    const int* __restrict__ nb_idx, const float* __restrict__ new_xyz,
    float* __restrict__ out_feat) {
  // 4 waves / block, one sampled point per wave.
  __shared__ float s_nfn[4][NSAMP][NFPITCH];   // 110.6 KB : per-point 32x212 tile
  __shared__ float s_acol[16 * NC];            // 13.6 KB  : a[:, nt*16 .. +16) transposed
  __shared__ float s_pool[4][NFPITCH];         // pooled features per point

  const int wid = threadIdx.x >> 5, lane = threadIdx.x & 31;
  const int p = blockIdx.x * 4 + wid;          // grid = 2048 -> p in [0, 8192)
  const int b = p >> 10, n = p & 1023;

  // ---- phase 1: build one nfn row per lane (lane == neighbor k) ----
  {
    const int k    = lane;
    const int fpsi = fps_idx[b * NFPS + n];
    const int nbi  = nb_idx[((size_t)b * NFPS + n) * NSAMP + k];
    const float* Xb = xyz + (size_t)b * NPTS * 3;
    const float* Q  = new_xyz + ((size_t)b * NFPS + n) * 3;
    const float gx = Q[0], gy = Q[1], gz = Q[2];
    float gnx = Xb[nbi * 3 + 0], gny = Xb[nbi * 3 + 1], gnz = Xb[nbi * 3 + 2];
    const float* Fb = features + (size_t)b * DFEAT * NPTS;
    float* row = &s_nfn[wid][k][0];

    float dsum = 0.f, df2 = 0.f;
    for (int c = 0; c < DFEAT; ++c) {
      float fc  = Fb[(size_t)c * NPTS + fpsi];
      float fnc = Fb[(size_t)c * NPTS + nbi];
      float dd  = fc - fnc;
      dsum += fabsf(dd);
      df2  += dd * dd;
      row[10 + c]  = fnc;    // fn
      row[77 + c]  = fc;     // pf (f part)
      row[144 + c] = dd;     // pf1 (f part)
    }
    const float fdist = expf(-dsum * (1.0f / DFEAT));
    gnx *= fdist; gny *= fdist; gnz *= fdist;

    // gmean over the 32 neighbors (cross-lane)
    float mx = gnx, my = gny, mz = gnz;
    for (int o = 16; o > 0; o >>= 1) {
      mx += __shfl_xor(mx, o); my += __shfl_xor(my, o); mz += __shfl_xor(mz, o);
    }
    mx *= (1.f / 32.f); my *= (1.f / 32.f); mz *= (1.f / 32.f);

    auto softmax3 = [](float& a, float& bq, float& c) {
      float m = fmaxf(a, fmaxf(bq, c));
      float ea = expf(a - m), eb = expf(bq - m), ec = expf(c - m);
      float s = ea + eb + ec;
      a = ea / s; bq = eb / s; c = ec / s;
    };
    float sgx = gx, sgy = gy, sgz = gz; softmax3(sgx, sgy, sgz);
    float smx = mx, smy = my, smz = mz; softmax3(smx, smy, smz);
    const float gdx = fabsf(smx - sgx), gdy = fabsf(smy - sgy), gdz = fabsf(smz - sgz);

    const float p1x = gx - gnx, p1y = gy - gny, p1z = gz - gnz;
    const float dg  = sqrtf(p1x * p1x + p1y * p1y + p1z * p1z);
    const float q1x = gx - gdx, q1y = gy - gdy, q1z = gz - gdz;
    df2 += q1x * q1x + q1y * q1y + q1z * q1z;
    const float df = sqrtf(df2);

    row[0] = gnx;  row[1] = gny;  row[2] = gnz;     // gn (scaled)
    row[3] = gx;   row[4] = gy;   row[5] = gz;      // pg
    row[6] = p1x;  row[7] = p1y;  row[8] = p1z;     // pg1
    row[9] = dg;                                    // dg
    row[74]  = gdx; row[75]  = gdy; row[76]  = gdz; // fn' tail: gd
    row[141] = gx;  row[142] = gy;  row[143] = gz;  // pf tail: g
    row[208] = q1x; row[209] = q1y; row[210] = q1z; // pf1 tail: g - gd
    row[211] = df;                                  // df
    row[212] = 0.f; row[213] = 0.f; row[214] = 0.f; row[215] = 0.f; // pad
  }
  __syncthreads();

  // ---- phase 2: e = leakyrelu(nfn @ a); softmax over k; pooled ----
  // A-frag (16x4 f32): lane L holds M=L%16, K = 2*(L/16)+{0,1}
  // B-frag (4x16 f32): lane L holds N=L%16, K = 2*(L/16)+{0,1}
  // C/D   (16x16 f32): lane L, vgpr v -> M = v + 8*(L/16), N = L%16
  const int r0 = lane & 15;
  const int kb = (lane >> 4) * 2;
  for (int nt = 0; nt < NTILES; ++nt) {
    __syncthreads();
    // stage a[:, nt*16 .. nt*16+16) transposed: s_acol[nn*212 + kk]
    for (int t = threadIdx.x; t < NC * 16; t += 128) {
      int nn = t / NC, kk = t - nn * NC;
      int col = nt * 16 + nn;
      s_acol[nn * NC + kk] = (col < NC) ? Amat[kk * NC + col] : 0.f;
    }
    __syncthreads();

    v8f acc0 = {0.f, 0.f, 0.f, 0.f, 0.f, 0.f, 0.f, 0.f};
    v8f acc1 = {0.f, 0.f, 0.f, 0.f, 0.f, 0.f, 0.f, 0.f};
    const float* arow0 = &s_nfn[wid][r0][kb];
    const float* arow1 = &s_nfn[wid][r0 + 16][kb];
    const float* brow  = &s_acol[r0 * NC + kb];
    for (int ks = 0; ks < KSTEPS; ++ks) {
      v2f a0 = *(const v2f*)(arow0 + 4 * ks);   // ds_load_b64 (even offsets)
      v2f a1 = *(const v2f*)(arow1 + 4 * ks);
      v2f bf = *(const v2f*)(brow + 4 * ks);
      acc0 = __builtin_amdgcn_wmma_f32_16x16x4_f32(false, a0, false, bf,
                                                   (short)0, acc0, false, false);
      acc1 = __builtin_amdgcn_wmma_f32_16x16x4_f32(false, a1, false, bf,
                                                   (short)0, acc1, false, false);
    }

    const int col = nt * 16 + r0;
    if (col < NC) {   // lanes L and L^16 diverge together (same r0)
      float e[16];
      float m = -3.4e38f;
      for (int v = 0; v < 8; ++v) {
        float x0 = acc0[v]; x0 = x0 > 0.f ? x0 : 0.2f * x0;   // leaky_relu
        float x1 = acc1[v]; x1 = x1 > 0.f ? x1 : 0.2f * x1;
        e[v] = x0; e[8 + v] = x1;
        m = fmaxf(m, fmaxf(x0, x1));
      }
      m = fmaxf(m, __shfl_xor(m, 16));          // 32-way max over k
      float s = 0.f;
      for (int v = 0; v < 16; ++v) { e[v] = expf(e[v] - m); s += e[v]; }
      s += __shfl_xor(s, 16);
      const float inv = 1.f / s;
      const int mbase = (lane >> 4) * 8;
      float ps = 0.f;
      for (int v = 0; v < 8; ++v) {
        ps += e[v]     * inv * s_nfn[wid][mbase + v][col];
        ps += e[8 + v] * inv * s_nfn[wid][mbase + v + 16][col];
      }
      ps += __shfl_xor(ps, 16);
      if (lane < 16) s_pool[wid][col] = ps;
    }
  }
  __syncthreads();

  // ---- phase 3: out = w1 @ pooled + b1, stored transposed (B, 128, n) ----
  const float* pool = s_pool[wid];
  for (int j = 0; j < 4; ++j) {
    const int o = lane + 32 * j;
    float acc = B1[o];
    const float* wrow = W1 + (size_t)o * NC;
    for (int c = 0; c < NC; ++c) acc += wrow[c] * pool[c];
    out_feat[((size_t)b * COUT + o) * NFPS + n] = acc;
  }
}

// ---------------------------------------------------------------------------
extern "C" void kernel_launch(void* const* d_in, const int* in_sizes, int n_in,
                              void* d_out, int out_size, void* d_ws, size_t ws_size,
                              hipStream_t stream) {
  (void)in_sizes; (void)n_in; (void)out_size; (void)ws_size;
  const float* xyz      = (const float*)d_in[0];   // (8,4096,3)
  const float* features = (const float*)d_in[1];   // (8,64,4096)
  const float* Amat     = (const float*)d_in[2];   // (212,212)
  const float* W1       = (const float*)d_in[3];   // (128,212)
  const float* B1       = (const float*)d_in[4];   // (128,)

  float* out      = (float*)d_out;
  float* new_xyz  = out;                            // (8,1024,3)
  float* out_feat = out + (size_t)BATCH * NFPS * 3; // (8,128,1024)

  int* fps = (int*)d_ws;                            // 8*1024 ints
  int* nb  = fps + BATCH * NFPS;                    // 8*1024*32 ints

  fps_kernel<<<BATCH, 1024, 0, stream>>>(xyz, fps);
  gather_newxyz_kernel<<<(BATCH * NFPS * 3 + 255) / 256, 256, 0, stream>>>(xyz, fps, new_xyz);
  ballquery_kernel<<<BATCH * 4, 256, 0, stream>>>(xyz, new_xyz, nb);
  attn_kernel<<<(BATCH * NFPS) / 4, 128, 0, stream>>>(xyz, features, Amat, W1, B1,
                                                      fps, nb, new_xyz, out_feat);
}